// TwoLayerGAT_20710332301831
// MI455X (gfx1250) — compile-verified
//
#include <hip/hip_runtime.h>
#include <cstdint>
#include <cstddef>

#define NEG_SLOPE 0.2f
#define ENC_NEG_INF 0x007FFFFFu   // order-encoded -inf

typedef __attribute__((ext_vector_type(2))) float v2f;
typedef __attribute__((ext_vector_type(8))) float v8f;

// Monotone float <-> uint mapping so float max == unsigned max (hardware atomic_max_u32).
__device__ __forceinline__ unsigned fenc(float f) {
    unsigned u = __float_as_uint(f);
    return (u & 0x80000000u) ? ~u : (u | 0x80000000u);
}
__device__ __forceinline__ float fdec(unsigned e) {
    unsigned u = (e & 0x80000000u) ? (e ^ 0x80000000u) : ~e;
    return __uint_as_float(u);
}

// ---------------- init: out buffers to bias, max buffers to -inf, denoms to 0 -------------
__global__ __launch_bounds__(256) void gat_init(
    float* __restrict__ out1, float* __restrict__ out2,
    const float* __restrict__ b1, const float* __restrict__ b2,
    unsigned* __restrict__ emax1, float* __restrict__ den1,
    unsigned* __restrict__ emax2, float* __restrict__ den2, int n) {
    long long i = (long long)blockIdx.x * blockDim.x + threadIdx.x;
    long long tot = (long long)n * 64;
    if (i < tot) {
        int j = (int)(i & 63);
        out1[i] = b1[j];
        out2[i] = b2[j];
    }
    if (i < (long long)n * 8) { emax1[i] = ENC_NEG_INF; den1[i] = 0.0f; }
    if (i < (long long)n)     { emax2[i] = ENC_NEG_INF; den2[i] = 0.0f; }
}

// ---------------- fp32 WMMA GEMM: C[M,64] = A[M,K] * B[K,64] ------------------------------
// One wave per 16-row strip; 4 16x16 n-tiles held in 4 v8f accumulators.
// V_WMMA_F32_16X16X4_F32 operand layout (ISA 7.12.2):
//   A 16x4: lanes 0-15 -> {K=k0,k0+1}, lanes 16-31 -> {K=k0+2,k0+3}, row = lane&15
//   B 4x16: VGPR0 lanes 0-15 row K=k0, lanes 16-31 row K=k0+2; VGPR1 rows k0+1 / k0+3
//   C/D 16x16: VGPR i -> M = i (lanes 0-15) or i+8 (lanes 16-31), col = lane&15
__global__ __launch_bounds__(256) void gemm_wmma_n64(
    const float* __restrict__ A, const float* __restrict__ B,
    float* __restrict__ C, int M, int K) {
    int wave = threadIdx.x >> 5;
    int lane = threadIdx.x & 31;
    int tile = blockIdx.x * (blockDim.x >> 5) + wave;
    int ntiles = (M + 15) >> 4;
    if (tile >= ntiles) return;            // wave-uniform exit: EXEC all-ones for WMMA

    int m0 = tile << 4;
    int r  = lane & 15;
    int kk = (lane >> 4) << 1;             // 0 or 2
    int rowA = m0 + r; if (rowA > M - 1) rowA = M - 1;
    const float* arow = A + (long long)rowA * K;

    v8f acc0 = {}, acc1 = {}, acc2 = {}, acc3 = {};
    for (int k0 = 0; k0 < K; k0 += 4) {
        v2f a = *(const v2f*)(arow + k0 + kk);
        const float* bp = B + (long long)(k0 + kk) * 64 + r;
        v2f b0 = { bp[0],  bp[64]  };
        v2f b1 = { bp[16], bp[80]  };
        v2f b2 = { bp[32], bp[96]  };
        v2f b3 = { bp[48], bp[112] };
        acc0 = __builtin_amdgcn_wmma_f32_16x16x4_f32(false, a, false, b0, (short)0, acc0, false, false);
        acc1 = __builtin_amdgcn_wmma_f32_16x16x4_f32(false, a, false, b1, (short)0, acc1, false, false);
        acc2 = __builtin_amdgcn_wmma_f32_16x16x4_f32(false, a, false, b2, (short)0, acc2, false, false);
        acc3 = __builtin_amdgcn_wmma_f32_16x16x4_f32(false, a, false, b3, (short)0, acc3, false, false);
    }

    int mt = m0 + ((lane >> 4) << 3);
    #pragma unroll
    for (int i = 0; i < 8; ++i) {
        int row = mt + i;
        if (row < M) {
            float* cp = C + (long long)row * 64 + r;
            cp[0]  = acc0[i];
            cp[16] = acc1[i];
            cp[32] = acc2[i];
            cp[48] = acc3[i];
        }
    }
}

// ---------------- per-node attention coefficients: el/er = <h[n,h,:], a_{l/r}[h,:]> --------
template <int HEADS, int FEATS>
__global__ __launch_bounds__(256) void attn_coef(
    const float* __restrict__ h, const float* __restrict__ al, const float* __restrict__ ar,
    float* __restrict__ el, float* __restrict__ er, int n) {
    int i = blockIdx.x * blockDim.x + threadIdx.x;     // over n*HEADS
    if (i >= n * HEADS) return;
    int hd   = i % HEADS;
    int node = i / HEADS;
    const float* hp  = h + (long long)node * (HEADS * FEATS) + hd * FEATS;
    const float* alp = al + hd * FEATS;
    const float* arp = ar + hd * FEATS;
    float sl = 0.f, sr = 0.f;
    #pragma unroll
    for (int f = 0; f < FEATS; ++f) { float v = hp[f]; sl += v * alp[f]; sr += v * arp[f]; }
    el[i] = sl; er[i] = sr;
}

// ---------------- edge pass 1: segment max over dst (atomic_max_u32 on encoded floats) ----
template <int HEADS>
__global__ __launch_bounds__(256) void edge_max(
    const int* __restrict__ src, const int* __restrict__ dst,
    const float* __restrict__ el, const float* __restrict__ er,
    unsigned* __restrict__ emax, int E) {
    int e = blockIdx.x * blockDim.x + threadIdx.x;
    if (e >= E) return;
    int s = src[e], d = dst[e];
    #pragma unroll
    for (int h = 0; h < HEADS; ++h) {
        float t = el[s * HEADS + h] + er[d * HEADS + h];
        t = t > 0.f ? t : NEG_SLOPE * t;
        atomicMax(&emax[d * HEADS + h], fenc(t));
    }
}

// ---------------- edge pass 2: denom = segment_sum(exp(e - max)) --------------------------
template <int HEADS>
__global__ __launch_bounds__(256) void edge_expsum(
    const int* __restrict__ src, const int* __restrict__ dst,
    const float* __restrict__ el, const float* __restrict__ er,
    const unsigned* __restrict__ emax, float* __restrict__ den, int E) {
    int e = blockIdx.x * blockDim.x + threadIdx.x;
    if (e >= E) return;
    int s = src[e], d = dst[e];
    #pragma unroll
    for (int h = 0; h < HEADS; ++h) {
        float t = el[s * HEADS + h] + er[d * HEADS + h];
        t = t > 0.f ? t : NEG_SLOPE * t;
        atomicAdd(&den[d * HEADS + h], __expf(t - fdec(emax[d * HEADS + h])));
    }
}

// ---------------- edge pass 3: out[dst] += alpha * h[src]; 64 lanes per edge --------------
template <int HEADS, int LOGF>
__global__ __launch_bounds__(256) void edge_aggregate(
    const int* __restrict__ src, const int* __restrict__ dst,
    const float* __restrict__ el, const float* __restrict__ er,
    const unsigned* __restrict__ emax, const float* __restrict__ den,
    const float* __restrict__ hsrc, float* __restrict__ out, long long total) {
    long long i = (long long)blockIdx.x * blockDim.x + threadIdx.x;
    if (i >= total) return;                 // total = E*64
    int e  = (int)(i >> 6);
    int j  = (int)(i & 63);
    int hd = j >> LOGF;
    int s = src[e], d = dst[e];
    float t = el[s * HEADS + hd] + er[d * HEADS + hd];
    t = t > 0.f ? t : NEG_SLOPE * t;
    float alpha = __expf(t - fdec(emax[d * HEADS + hd])) / den[d * HEADS + hd];
    atomicAdd(&out[(long long)d * 64 + j], alpha * hsrc[(long long)s * 64 + j]);
}

// ---------------- ELU in place -------------------------------------------------------------
__global__ __launch_bounds__(256) void elu_inplace(float* __restrict__ x, long long tot) {
    long long i = (long long)blockIdx.x * blockDim.x + threadIdx.x;
    if (i >= tot) return;
    float v = x[i];
    x[i] = v > 0.f ? v : (__expf(v) - 1.f);
}

// ---------------- host driver --------------------------------------------------------------
extern "C" void kernel_launch(void* const* d_in, const int* in_sizes, int n_in,
                              void* d_out, int out_size, void* d_ws, size_t ws_size,
                              hipStream_t stream) {
    const float* x    = (const float*)d_in[0];
    const int*   esrc = (const int*)  d_in[1];
    const int*   edst = (const int*)  d_in[2];
    const float* W1   = (const float*)d_in[3];
    const float* al1  = (const float*)d_in[4];
    const float* ar1  = (const float*)d_in[5];
    const float* b1   = (const float*)d_in[6];
    const float* W2   = (const float*)d_in[7];
    const float* al2  = (const float*)d_in[8];
    const float* ar2  = (const float*)d_in[9];
    const float* b2   = (const float*)d_in[10];
    float* out = (float*)d_out;

    const int FIN = 128;
    int n = in_sizes[0] / FIN;
    int E = in_sizes[1];

    // Workspace carve (256B aligned). Total ~66 MB. h1 buffer is reused for layer-2 feats.
    char* w = (char*)d_ws;
    auto carve = [&](size_t bytes) -> void* {
        void* p = (void*)w;
        w += (bytes + 255) & ~(size_t)255;
        return p;
    };
    float*    h1    = (float*)   carve((size_t)n * 64 * 4);  // x@W1, then reused as h@W2
    float*    out1  = (float*)   carve((size_t)n * 64 * 4);  // layer1 agg (+bias), ELU in place
    float*    el1   = (float*)   carve((size_t)n * 8 * 4);
    float*    er1   = (float*)   carve((size_t)n * 8 * 4);
    unsigned* emax1 = (unsigned*)carve((size_t)n * 8 * 4);
    float*    den1  = (float*)   carve((size_t)n * 8 * 4);
    float*    el2   = (float*)   carve((size_t)n * 4);
    float*    er2   = (float*)   carve((size_t)n * 4);
    unsigned* emax2 = (unsigned*)carve((size_t)n * 4);
    float*    den2  = (float*)   carve((size_t)n * 4);

    const int T = 256;
    long long tot64 = (long long)n * 64;
    long long etot  = (long long)E * 64;
    dim3 blk(T);
    unsigned gN64  = (unsigned)((tot64 + T - 1) / T);
    unsigned gE    = (unsigned)((E + T - 1) / T);
    unsigned gE64  = (unsigned)((etot + T - 1) / T);
    int waves      = (n + 15) >> 4;
    unsigned gGemm = (unsigned)((waves + 7) >> 3);           // 8 waves per 256-thread block

    gat_init<<<dim3(gN64), blk, 0, stream>>>(out1, out, b1, b2, emax1, den1, emax2, den2, n);

    // ---- layer 1 (8 heads x 8 feats) ----
    gemm_wmma_n64<<<dim3(gGemm), blk, 0, stream>>>(x, W1, h1, n, 128);
    attn_coef<8, 8><<<dim3((unsigned)((n * 8 + T - 1) / T)), blk, 0, stream>>>(h1, al1, ar1, el1, er1, n);
    edge_max<8>   <<<dim3(gE),   blk, 0, stream>>>(esrc, edst, el1, er1, emax1, E);
    edge_expsum<8><<<dim3(gE),   blk, 0, stream>>>(esrc, edst, el1, er1, emax1, den1, E);
    edge_aggregate<8, 3><<<dim3(gE64), blk, 0, stream>>>(esrc, edst, el1, er1, emax1, den1, h1, out1, etot);
    elu_inplace<<<dim3(gN64), blk, 0, stream>>>(out1, tot64);

    // ---- layer 2 (1 head x 64 feats) ----
    float* h2 = h1;                                          // h1 dead; reuse buffer
    gemm_wmma_n64<<<dim3(gGemm), blk, 0, stream>>>(out1, W2, h2, n, 64);
    attn_coef<1, 64><<<dim3((unsigned)((n + T - 1) / T)), blk, 0, stream>>>(h2, al2, ar2, el2, er2, n);
    edge_max<1>   <<<dim3(gE),   blk, 0, stream>>>(esrc, edst, el2, er2, emax2, E);
    edge_expsum<1><<<dim3(gE),   blk, 0, stream>>>(esrc, edst, el2, er2, emax2, den2, E);
    edge_aggregate<1, 6><<<dim3(gE64), blk, 0, stream>>>(esrc, edst, el2, er2, emax2, den2, h2, out, etot);
}